// GMDTW_CUDA_57475252355414
// MI455X (gfx1250) — compile-verified
//
#include <hip/hip_runtime.h>

// ---------------------------------------------------------------------------
// Soft-DTW (GMDTW) for M=2048, D=64, gamma=1.0 on gfx1250 (MI455X).
//   Phase 1: C[i,j] = ||x1_i||^2 + ||x2_j||^2 - 2 * <x1_i, x2_j>
//            via exact-fp32 V_WMMA_F32_16X16X4_F32 (16x16 tile per wave32).
//   Phase 2: antidiagonal wavefront DP in one 1024-thread workgroup,
//            3 rotating LDS diagonals, one barrier per diagonal.
// ---------------------------------------------------------------------------

typedef float v2f __attribute__((ext_vector_type(2)));
typedef float v8f __attribute__((ext_vector_type(8)));

#define MDIM 2048
#define DDIM 64

// --- Kernel 0: per-row sum of squares for x1 (-> sq1) and x2 (-> sq2) ------
__global__ void gmdtw_sumsq(const float* __restrict__ x1,
                            const float* __restrict__ x2,
                            float* __restrict__ sq1,
                            float* __restrict__ sq2) {
  int idx = blockIdx.x * blockDim.x + threadIdx.x;
  if (idx >= 2 * MDIM) return;
  const float* src = (idx < MDIM) ? (x1 + (size_t)idx * DDIM)
                                  : (x2 + (size_t)(idx - MDIM) * DDIM);
  float s = 0.0f;
#pragma unroll
  for (int k = 0; k < DDIM; k += 4) {
    float4 v = *reinterpret_cast<const float4*>(src + k);
    s = fmaf(v.x, v.x, s);
    s = fmaf(v.y, v.y, s);
    s = fmaf(v.z, v.z, s);
    s = fmaf(v.w, v.w, s);
  }
  if (idx < MDIM) sq1[idx] = s;
  else            sq2[idx - MDIM] = s;
}

// --- Kernel 1: cost matrix via fp32 WMMA -----------------------------------
// One wave32 per 16x16 tile of C; 128x128 tiles; 8 waves per 256-thread block.
// A-matrix (16x4 f32) lane layout: lane<16 holds A[lane][k0..k0+1],
// lane>=16 holds A[lane-16][k0+2..k0+3]; B (4x16) mirrors it with N<->M,
// so both A (x1 rows) and B (x2 rows, since B = x2^T) use identical
// float2 addressing from row-major storage.
__global__ void gmdtw_cost_wmma(const float* __restrict__ x1,
                                const float* __restrict__ x2,
                                const float* __restrict__ sq1,
                                const float* __restrict__ sq2,
                                float* __restrict__ C) {
  const int wave = threadIdx.x >> 5;
  const int lane = threadIdx.x & 31;
  const int tile = blockIdx.x * 8 + wave;      // 0 .. 16383
  const int tr = tile >> 7;                    // tile row 0..127
  const int tc = tile & 127;                   // tile col 0..127
  const int rowBase = tr * 16;
  const int colBase = tc * 16;

  const int lm = lane & 15;                    // M (for A) / N (for B)
  const int kh = (lane >> 4) << 1;             // K half-offset: 0 or 2

  const float* arow = x1 + (size_t)(rowBase + lm) * DDIM + kh;
  const float* brow = x2 + (size_t)(colBase + lm) * DDIM + kh;

  v8f acc = {};
#pragma unroll
  for (int kk = 0; kk < DDIM; kk += 4) {
    v2f a = *reinterpret_cast<const v2f*>(arow + kk);
    v2f b = *reinterpret_cast<const v2f*>(brow + kk);
    // (neg_a, A, neg_b, B, c_mod, C, reuse_a, reuse_b)
    acc = __builtin_amdgcn_wmma_f32_16x16x4_f32(false, a, false, b,
                                                (short)0, acc, false, false);
  }

  // C/D layout: VGPR r, lanes 0-15 -> M=r, N=lane; lanes 16-31 -> M=r+8.
  const int col = colBase + lm;
  const float s2 = sq2[col];
  const int rbase = rowBase + ((lane >> 4) << 3);
#pragma unroll
  for (int r = 0; r < 8; ++r) {
    const int row = rbase + r;
    C[(size_t)row * MDIM + col] = sq1[row] + s2 - 2.0f * acc[r];
  }
}

// --- Kernel 2: wavefront soft-DTW DP (single workgroup, gamma = 1) ---------
__global__ __launch_bounds__(1024) void gmdtw_dp(const float* __restrict__ C,
                                                 float* __restrict__ out) {
  __shared__ float buf[3][MDIM];               // 24 KB of 320 KB WGP LDS
  const float INF = __builtin_inff();
  const int t = threadIdx.x;

#pragma unroll
  for (int b = 0; b < 3; ++b) {
    buf[b][t] = INF;
    buf[b][t + 1024] = INF;
  }
  __syncthreads();

  for (int p = 0; p < 2 * MDIM - 1; ++p) {
    const int cur = p % 3;                     // overwrites diagonal p-3
    const int b1  = (p + 2) % 3;               // diagonal p-1
    const int b2  = (p + 1) % 3;               // diagonal p-2
#pragma unroll
    for (int h = 0; h < 2; ++h) {
      const int i = t + h * 1024;              // fixed row per thread
      const int j = p - i;
      float val = INF;
      if (j >= 0 && j < MDIM) {
        const float left = buf[b1][i];                               // R[i, j-1]
        const float up   = (i == 0) ? INF : buf[b1][i - 1];          // R[i-1, j]
        const float diag = (i == 0) ? ((p == 0) ? 0.0f : INF)
                                    : buf[b2][i - 1];                // R[i-1, j-1]
        // stable softmin_1(left, up, diag)
        const float a = -left, b = -up, c = -diag;
        float rmax = fmaxf(a, fmaxf(b, c));
        const float fm = (rmax > -INF && rmax < INF) ? rmax : 0.0f;
        const float s = __expf(a - fm) + __expf(b - fm) + __expf(c - fm);
        const float sm = (s > 0.0f) ? -(__logf(s) + fm) : INF;
        val = C[(size_t)i * MDIM + j] + sm;
        if (p == 2 * MDIM - 2 && i == MDIM - 1) out[0] = val;        // R[m-1,m-1]
      }
      buf[cur][i] = val;
    }
    __syncthreads();
  }
}

// ---------------------------------------------------------------------------
extern "C" void kernel_launch(void* const* d_in, const int* in_sizes, int n_in,
                              void* d_out, int out_size, void* d_ws, size_t ws_size,
                              hipStream_t stream) {
  (void)in_sizes; (void)n_in; (void)out_size; (void)ws_size;
  const float* x1 = (const float*)d_in[0];
  const float* x2 = (const float*)d_in[1];

  // workspace layout: C (M*M floats) | sq1 (M floats) | sq2 (M floats)
  float* C   = (float*)d_ws;
  float* sq1 = C + (size_t)MDIM * MDIM;
  float* sq2 = sq1 + MDIM;

  gmdtw_sumsq<<<(2 * MDIM + 255) / 256, 256, 0, stream>>>(x1, x2, sq1, sq2);

  // 128*128 = 16384 tiles, 8 waves (tiles) per 256-thread block -> 2048 blocks
  gmdtw_cost_wmma<<<(128 * 128) / 8, 256, 0, stream>>>(x1, x2, sq1, sq2, C);

  gmdtw_dp<<<1, 1024, 0, stream>>>(C, (float*)d_out);
}